// SparseEmbedding_19464791786180
// MI455X (gfx1250) — compile-verified
//
#include <hip/hip_runtime.h>

// Embedding-bag: out[b,:] = sum_n vals[b,n] * kernel[idx[b,n],:] + bias
// B=4096, NNZ=32, VOCAB=1e6, DIM=64.
// Roofline: ~35 MB of random 256 B row gathers @ 23.3 TB/s => ~1.5 us floor;
// FLOPs (16.8 M) are free. One wave32 per batch row; the per-row [1x32]@[32x64]
// reduction runs on V_WMMA_F32_16X16X4_F32 with the vals vector replicated
// across all 16 M rows (exact f32, every D row holds the answer).
// v3: straight-line v_readlane broadcasts (hoisted OUT of the ternaries so the
// convergent readlanes stay in uniform control flow -> no exec-mask branching),
// then v_cndmask selects between the uniform scalars. No LDS, no DScnt waits.

#define EB_BATCH 4096
#define EB_NNZ   32
#define EB_DIM   64

typedef float v2f __attribute__((ext_vector_type(2)));
typedef float v8f __attribute__((ext_vector_type(8)));

static __device__ __forceinline__ float readlane_f(float v, int l) {
    return __int_as_float(__builtin_amdgcn_readlane(__float_as_int(v), l));
}

__global__ __launch_bounds__(256) void SparseEmbedding_wmma_kernel(
    const int*   __restrict__ idx,     // [B, NNZ]
    const float* __restrict__ vals,    // [B, NNZ]
    const float* __restrict__ table,   // [VOCAB, DIM]
    const float* __restrict__ bias,    // [DIM]
    float*       __restrict__ out)     // [B, DIM]
{
    const int lane = threadIdx.x & 31;
    const int wave = threadIdx.x >> 5;
    const int b    = blockIdx.x * 8 + wave;    // one wave per batch row

    // Lane L caches nnz slot L of this row (one coalesced b32 load each).
    const int   my_idx = idx [b * EB_NNZ + lane];
    const float my_val = vals[b * EB_NNZ + lane];

    // f32 WMMA A/B striping: VGPR0 holds K=0 (lanes 0-15) / K=2 (lanes 16-31),
    // VGPR1 holds K=1 / K=3.
    const bool lo    = lane < 16;
    const int  nlane = lane & 15;              // N position within a 16-dim chunk

    v8f acc0 = {}, acc1 = {}, acc2 = {}, acc3 = {};

    #pragma unroll
    for (int j = 0; j < 8; ++j) {              // K-steps of 4 over NNZ=32
        const int k0 = 4 * j;

        // Unconditional constant-lane readlanes (uniform SGPR results,
        // straight-line code). Selects below become v_cndmask, not branches.
        const int   i0 = __builtin_amdgcn_readlane(my_idx, k0);
        const int   i1 = __builtin_amdgcn_readlane(my_idx, k0 + 1);
        const int   i2 = __builtin_amdgcn_readlane(my_idx, k0 + 2);
        const int   i3 = __builtin_amdgcn_readlane(my_idx, k0 + 3);
        const float v0 = readlane_f(my_val, k0);
        const float v1 = readlane_f(my_val, k0 + 1);
        const float v2 = readlane_f(my_val, k0 + 2);
        const float v3 = readlane_f(my_val, k0 + 3);

        const int r0 = lo ? i0 : i2;           // row for this lane, VGPR0 slot
        const int r1 = lo ? i1 : i3;           // row for this lane, VGPR1 slot

        v2f a;                                 // A rows replicated with vals
        a.x = lo ? v0 : v2;
        a.y = lo ? v1 : v3;

        const unsigned o0 = (unsigned)r0 * EB_DIM + nlane;  // 32-bit offsets
        const unsigned o1 = (unsigned)r1 * EB_DIM + nlane;  // (256 MB table)

        // B tiles for the four 16-dim chunks; each half-wave loads a
        // contiguous 64 B segment of a gathered row -> fully coalesced,
        // and one wave consumes each 256 B table row exactly once.
        v2f b0, b1, b2, b3;
        b0.x = table[o0];       b0.y = table[o1];
        b1.x = table[o0 + 16];  b1.y = table[o1 + 16];
        b2.x = table[o0 + 32];  b2.y = table[o1 + 32];
        b3.x = table[o0 + 48];  b3.y = table[o1 + 48];

        acc0 = __builtin_amdgcn_wmma_f32_16x16x4_f32(false, a, false, b0,
                                                     (short)0, acc0, false, false);
        acc1 = __builtin_amdgcn_wmma_f32_16x16x4_f32(false, a, false, b1,
                                                     (short)0, acc1, false, false);
        acc2 = __builtin_amdgcn_wmma_f32_16x16x4_f32(false, a, false, b2,
                                                     (short)0, acc2, false, false);
        acc3 = __builtin_amdgcn_wmma_f32_16x16x4_f32(false, a, false, b3,
                                                     (short)0, acc3, false, false);
    }

    // D layout: VGPR0 = (M=0, N=lane) in lanes 0-15 and (M=8, N=lane-16) in
    // lanes 16-31. All M rows are identical, so EVERY lane's element 0 holds
    // out[b, chunk*16 + (lane&15)] -- split the stores across both halves.
    float* o = out + (size_t)b * EB_DIM;
    if (lo) {
        o[nlane]      = acc0[0] + bias[nlane];
        o[nlane + 16] = acc1[0] + bias[nlane + 16];
    } else {
        o[nlane + 32] = acc2[0] + bias[nlane + 32];
        o[nlane + 48] = acc3[0] + bias[nlane + 48];
    }
}

extern "C" void kernel_launch(void* const* d_in, const int* in_sizes, int n_in,
                              void* d_out, int out_size, void* d_ws, size_t ws_size,
                              hipStream_t stream) {
    const int*   idx   = (const int*)  d_in[0];   // [4096, 32] int32
    const float* vals  = (const float*)d_in[1];   // [4096, 32] f32
    const float* table = (const float*)d_in[2];   // [1000000, 64] f32
    const float* bias  = (const float*)d_in[3];   // [64] f32
    float*       out   = (float*)d_out;           // [4096, 64] f32

    // 8 waves (256 threads) per block, one wave per batch row -> 512 blocks.
    dim3 grid(EB_BATCH / 8);
    dim3 block(256);
    SparseEmbedding_wmma_kernel<<<grid, block, 0, stream>>>(idx, vals, table, bias, out);
}